// SinkhornEvalKBestAbs_14654428414012
// MI455X (gfx1250) — compile-verified
//
#include <hip/hip_runtime.h>
#include <math.h>

// SinkhornEvalKBestAbs on MI455X (gfx1250, wave32).
// N=65536 rows; per row: max_k |x[n,k,:]. y[n,:]| over K=1024, D=4,
// then 2*acos(clamped max), averaged over n.  HBM-bound: 1 GiB of x streamed
// once -> ~46 us floor at 23.3 TB/s.  Dot products are routed through
// V_WMMA_F32_16X16X4_F32 (full f32 precision, required because acos is
// ill-conditioned near |c|->1 and the clamp threshold is 1e-7 from 1.0).
// min over distances == 2*acos(max |cosine|) since acos is monotone
// decreasing and the clamp preserves ordering; acos evaluated once per n.

typedef __attribute__((ext_vector_type(2))) float v2f;
typedef __attribute__((ext_vector_type(8))) float v8f;

#define NN 65536
#define KK 1024
#define WAVES_PER_BLOCK 8
#define NBLOCKS (NN / WAVES_PER_BLOCK) /* 8192 */
#define UPPER_CLAMP 0.9999999f

__global__ __launch_bounds__(256) void
minarc_wmma_kernel(const float* __restrict__ y, const float* __restrict__ x,
                   float* __restrict__ partial) {
  const int lane = threadIdx.x & 31;
  const int wave = threadIdx.x >> 5;
  const int n = blockIdx.x * WAVES_PER_BLOCK + wave;

  const float* xn = x + (size_t)n * (KK * 4);

  // y[n] : 4 floats, 16B aligned; broadcast load (hits L0/L2, tiny footprint).
  const float4 yv = *(const float4*)(y + (size_t)n * 4);

  // B matrix (4x16 f32): every column j equals y[n], so D[i][j] = x[k_i].y[n]
  // for ALL i,j.  Fragment layout assumption (mirrors the documented f32 A
  // layout's even/odd K split across lane halves):
  //   VGPR0: lanes 0-15 -> row K=0, lanes 16-31 -> row K=2
  //   VGPR1: lanes 0-15 -> row K=1, lanes 16-31 -> row K=3
  v2f b;
  b.x = (lane < 16) ? yv.x : yv.z;
  b.y = (lane < 16) ? yv.y : yv.w;

  // A matrix (16x4 f32), documented layout:
  //   VGPR0: lanes 0-15 -> (M=lane, K=0); lanes 16-31 -> (M=lane-16, K=2)
  //   VGPR1: lanes 0-15 -> (M=lane, K=1); lanes 16-31 -> (M=lane-16, K=3)
  // Per-lane b64 load; the 32 lanes of the wave together cover one contiguous
  // 256-byte chunk = 16 consecutive k-rows of x[n].
  const int laneOff = (lane & 15) * 4 + (lane >> 4) * 2; // in floats

  float m = 0.0f;
#pragma unroll 4
  for (int c = 0; c < KK / 16; ++c) {
    // stream-once data: non-temporal load (TH=NT), don't pollute caches
    v2f a = __builtin_nontemporal_load((const v2f*)(xn + c * 64 + laneOff));
    v8f cz = {};
    // 8 args: (neg_a, A, neg_b, B, c_mod, C, reuse_a, reuse_b)
    v8f d = __builtin_amdgcn_wmma_f32_16x16x4_f32(
        false, a, false, b, (short)0, cz, false, false);
    // All 16 columns of D are identical, so the per-lane max over the 8 D
    // VGPRs covers M=0..7 (lanes 0-15) or M=8..15 (lanes 16-31).
#pragma unroll
    for (int i = 0; i < 8; ++i) m = fmaxf(m, fabsf(d[i]));
  }

  // Columns identical -> one SWAPX16 lane exchange completes the wave max
  // (ds_swizzle group-of-32: and=0x1f, or=0, xor=0x10 -> offset 0x401F).
  {
    float other =
        __int_as_float(__builtin_amdgcn_ds_swizzle(__float_as_int(m), 0x401F));
    m = fmaxf(m, other);
  }

  __shared__ float smem[WAVES_PER_BLOCK];
  if (lane == 0) {
    float c = (m > UPPER_CLAMP) ? 1.0f : m; // acos(1)=0 == min distance
    smem[wave] = 2.0f * acosf(c);
  }
  __syncthreads();
  if (threadIdx.x == 0) {
    float s = 0.0f;
#pragma unroll
    for (int w = 0; w < WAVES_PER_BLOCK; ++w) s += smem[w];
    partial[blockIdx.x] = s; // fixed order -> deterministic
  }
}

__global__ __launch_bounds__(256) void
reduce_kernel(const float* __restrict__ partial, float* __restrict__ out) {
  __shared__ float smem[256];
  float s = 0.0f;
  for (int i = threadIdx.x; i < NBLOCKS; i += 256) s += partial[i];
  smem[threadIdx.x] = s;
  __syncthreads();
  for (int off = 128; off > 0; off >>= 1) {
    if ((int)threadIdx.x < off) smem[threadIdx.x] += smem[threadIdx.x + off];
    __syncthreads();
  }
  if (threadIdx.x == 0) out[0] = smem[0] / (float)NN;
}

extern "C" void kernel_launch(void* const* d_in, const int* in_sizes, int n_in,
                              void* d_out, int out_size, void* d_ws,
                              size_t ws_size, hipStream_t stream) {
  // setup_inputs order: y (N*1*4 f32), w_y (N*K f32, UNUSED), x (N*K*4 f32)
  const float* y = (const float*)d_in[0];
  const float* x = (const float*)d_in[2];
  float* out = (float*)d_out;
  float* partial = (float*)d_ws; // needs NBLOCKS*4 = 32 KB of scratch

  minarc_wmma_kernel<<<NBLOCKS, 256, 0, stream>>>(y, x, partial);
  reduce_kernel<<<1, 256, 0, stream>>>(partial, out);
}